// micro_61899068670574
// MI455X (gfx1250) — compile-verified
//
#include <hip/hip_runtime.h>

typedef __attribute__((ext_vector_type(2))) float v2f;
typedef __attribute__((ext_vector_type(4))) float v4f;
typedef __attribute__((ext_vector_type(8))) float v8f;

#define DCOLS 128
#define LDSW 132                    // padded row stride (words); 132 % 64 == 4 -> conflict-free strided reads
#define ROWS_PER_WAVE 16
#define WAVES_PER_BLOCK 8
#define ROWS_PER_BLOCK (ROWS_PER_WAVE * WAVES_PER_BLOCK)
#define WAVE_LDS_FLOATS (2 * ROWS_PER_WAVE * LDSW + 16)   // clear tile + cor tile + 16 dots

__global__ __launch_bounds__(256) void fuse_gram_kernel(
    const float* __restrict__ feat_clear,
    const float* __restrict__ feat_cor,
    const float* __restrict__ tptr,
    float* __restrict__ out,
    int nrows)
{
    extern __shared__ float smem[];

    const int lane = threadIdx.x & 31;
    const int wv   = threadIdx.x >> 5;

    float* sClear = smem + wv * WAVE_LDS_FLOATS;
    float* sCor   = sClear + ROWS_PER_WAVE * LDSW;
    float* sDots  = sCor   + ROWS_PER_WAVE * LDSW;

    const long rowBase = (long)blockIdx.x * ROWS_PER_BLOCK + (long)wv * ROWS_PER_WAVE;
    if (rowBase >= nrows) return;

    const float t   = *tptr;            // uniform scalar -> s_load path
    const float u   = 1.0f - t;
    const float t2c = t * t + u * u;
    const float tt2 = 2.0f * t * u;

    const float* gA = feat_clear + rowBase * DCOLS;
    const float* gB = feat_cor   + rowBase * DCOLS;
    float*       gO = out        + rowBase * DCOLS;

    // ---- Phase 1: stage 16x128 tiles of both inputs into LDS (coalesced b128) ----
    #pragma unroll
    for (int r = 0; r < ROWS_PER_WAVE; ++r) {
        v4f a = *(const v4f*)(gA + r * DCOLS + lane * 4);
        v4f b = *(const v4f*)(gB + r * DCOLS + lane * 4);
        *(v4f*)(sClear + r * LDSW + lane * 4) = a;
        *(v4f*)(sCor   + r * LDSW + lane * 4) = b;
    }
    asm volatile("s_wait_dscnt 0" ::: "memory");

    // ---- Phase 2: Gram tile = clear_tile (16x128) x cor_tile^T (128x16) via f32 WMMA ----
    // A fragment (16x4 f32): lane holds M = lane&15, K = k0 + 2*(lane>>4) + {0,1}
    // B fragment (4x16 f32): lane holds N = lane&15, K = k0 + 2*(lane>>4) + {0,1}
    const int half = lane >> 4;
    const int mr   = lane & 15;
    v8f c = {};
    #pragma unroll
    for (int k0 = 0; k0 < DCOLS; k0 += 4) {
        v2f af = *(const v2f*)(sClear + mr * LDSW + k0 + 2 * half);
        v2f bf = *(const v2f*)(sCor   + mr * LDSW + k0 + 2 * half);
        c = __builtin_amdgcn_wmma_f32_16x16x4_f32(
                /*neg_a=*/false, af, /*neg_b=*/false, bf,
                /*c_mod=*/(short)0, c, /*reuse_a=*/false, /*reuse_b=*/false);
    }

    // ---- Phase 3: extract diagonal of 16x16 f32 C/D tile ----
    // VGPR r: lanes 0-15 -> (M=r, N=lane); lanes 16-31 -> (M=r+8, N=lane-16)
    // diag[m] (m<8)  at lane m,    c[m]
    // diag[m] (m>=8) at lane m+16, c[m-8]
    {
        int ci = (lane < 8) ? lane : (lane - 24);
        float dv = c[0];
        dv = (ci == 1) ? c[1] : dv;
        dv = (ci == 2) ? c[2] : dv;
        dv = (ci == 3) ? c[3] : dv;
        dv = (ci == 4) ? c[4] : dv;
        dv = (ci == 5) ? c[5] : dv;
        dv = (ci == 6) ? c[6] : dv;
        dv = (ci == 7) ? c[7] : dv;
        if (lane < 8)        sDots[lane]      = dv;
        else if (lane >= 24) sDots[lane - 16] = dv;
    }
    asm volatile("s_wait_dscnt 0" ::: "memory");

    // ---- Phase 4: numerator / denominator, coalesced b128 stores ----
    #pragma unroll
    for (int r = 0; r < ROWS_PER_WAVE; ++r) {
        float dot  = sDots[r];                       // uniform broadcast read
        float inv  = 1.0f / sqrtf(fmaf(tt2, dot, t2c));
        v4f a = *(const v4f*)(sClear + r * LDSW + lane * 4);
        v4f b = *(const v4f*)(sCor   + r * LDSW + lane * 4);
        v4f o;
        o.x = fmaf(t, b.x, u * a.x) * inv;
        o.y = fmaf(t, b.y, u * a.y) * inv;
        o.z = fmaf(t, b.z, u * a.z) * inv;
        o.w = fmaf(t, b.w, u * a.w) * inv;
        *(v4f*)(gO + r * DCOLS + lane * 4) = o;
    }
}

extern "C" void kernel_launch(void* const* d_in, const int* in_sizes, int n_in,
                              void* d_out, int out_size, void* d_ws, size_t ws_size,
                              hipStream_t stream) {
    const float* feat_clear = (const float*)d_in[0];
    const float* feat_cor   = (const float*)d_in[1];
    const float* tptr       = (const float*)d_in[2];
    float*       out        = (float*)d_out;

    const int nrows = in_sizes[0] / DCOLS;            // 1048576
    const int grid  = (nrows + ROWS_PER_BLOCK - 1) / ROWS_PER_BLOCK;
    const size_t shmem = (size_t)WAVES_PER_BLOCK * WAVE_LDS_FLOATS * sizeof(float);

    fuse_gram_kernel<<<grid, 256, shmem, stream>>>(feat_clear, feat_cor, tptr, out, nrows);
}